// BilinearInterpolation__13443247637073
// MI455X (gfx1250) — compile-verified
//
#include <hip/hip_runtime.h>
#include <hip/hip_bf16.h>

// Bilinear warp (grid sample with additive displacement field), B=64, H=W=512, C=1.
// Pure bandwidth-bound gather: ~268 MB of HBM traffic -> ~11.5 us floor at 23.3 TB/s.
// Strategy: b128 NT streaming loads/stores for dvfs/out, temporal scalar gathers for
// imgs (fits in 192MB L2), gfx1250 global_prefetch_b8 of the identity-grid rows to
// warm the cachelines the data-dependent gathers will hit.

typedef float v4f __attribute__((ext_vector_type(4)));

#define BK_W 512
#define BK_H 512

__global__ __launch_bounds__(256) void bilerp_warp_kernel(
    const float* __restrict__ imgs,   // (B, H, W)   C==1
    const float* __restrict__ dvfs,   // (B, H, W, 2)
    float* __restrict__ out)          // (B, H, W)
{
    // Each thread computes 4 consecutive x-pixels of one row.
    const int gid = blockIdx.x * blockDim.x + threadIdx.x;   // 0 .. B*H*W/4 - 1
    const int xg     = (gid & (BK_W / 4 - 1)) << 2;          // x start of 4-pixel group
    const int rowIdx = gid >> 7;                             // b*H + y   (W/4 = 128)
    const int y      = rowIdx & (BK_H - 1);
    const int b      = rowIdx >> 9;                          // H = 2^9

    const float* img = imgs + ((long)b << 18);               // b * H * W

    // CDNA5 prefetch: displacements are ~N(0,1), so the gathered neighborhood is
    // within a few pixels of the identity grid. Warm rows y and y+1 at this column.
    {
        const int yp = (y + 1 < BK_H) ? (y + 1) : (BK_H - 1);
        __builtin_prefetch(img + ((long)y  << 9) + xg, 0, 1); // global_prefetch_b8
        __builtin_prefetch(img + ((long)yp << 9) + xg, 0, 1);
    }

    // Stream 8 floats (4 pixels x float2) of the displacement field: two NT b128 loads.
    const v4f* dptr = (const v4f*)(dvfs + ((long)gid << 3));
    const v4f  d0   = __builtin_nontemporal_load(dptr);
    const v4f  d1   = __builtin_nontemporal_load(dptr + 1);

    const float dx[4] = { d0.x, d0.z, d1.x, d1.z };
    const float dy[4] = { d0.y, d0.w, d1.y, d1.w };

    v4f res;
#pragma unroll
    for (int i = 0; i < 4; ++i) {
        const float xs = dx[i] + (float)(xg + i);
        const float ys = dy[i] + (float)y;

        // truncation toward zero == jnp astype(int32) semantics
        const int x0 = (int)xs;
        const int y0 = (int)ys;
        const int x1 = x0 + 1;
        const int y1 = y0 + 1;

        const int x0c = min(max(x0, 0), BK_W - 1);
        const int x1c = min(max(x1, 0), BK_W - 1);
        const int y0c = min(max(y0, 0), BK_H - 1);
        const int y1c = min(max(y1, 0), BK_H - 1);

        // weights from CLIPPED corner coords (matches reference exactly)
        const float x0f = (float)x0c, x1f = (float)x1c;
        const float y0f = (float)y0c, y1f = (float)y1c;
        const float wa = (x1f - xs) * (y1f - ys);
        const float wb = (x1f - xs) * (ys - y0f);
        const float wc = (xs - x0f) * (y1f - ys);
        const float wd = (xs - x0f) * (ys - y0f);

        // 4 temporal gathers (neighbor threads / corners share cachelines; L2-resident)
        const float* r0 = img + ((long)y0c << 9);
        const float* r1 = img + ((long)y1c << 9);
        const float pa = r0[x0c];
        const float pb = r1[x0c];
        const float pc = r0[x1c];
        const float pd = r1[x1c];

        res[i] = wa * pa + wb * pb + wc * pc + wd * pd;
    }

    // Output is write-once: NT b128 store, keep L2 for imgs.
    __builtin_nontemporal_store(res, (v4f*)(out + ((long)gid << 2)));
}

extern "C" void kernel_launch(void* const* d_in, const int* in_sizes, int n_in,
                              void* d_out, int out_size, void* d_ws, size_t ws_size,
                              hipStream_t stream) {
    const float* imgs = (const float*)d_in[0];  // (64,512,512,1) float32
    const float* dvfs = (const float*)d_in[1];  // (64,512,512,2) float32
    float* out = (float*)d_out;                 // (64,512,512,1) float32

    // B*H*W/4 threads, 256 threads/block (8 wave32 waves)
    const int total_groups = (64 * 512 * 512) / 4;      // 4,194,304
    const int block = 256;
    const int grid  = total_groups / block;             // 16,384
    bilerp_warp_kernel<<<grid, block, 0, stream>>>(imgs, dvfs, out);
}